// VNEmbedding_46308337385485
// MI455X (gfx1250) — compile-verified
//
#include <hip/hip_runtime.h>

// CDNA5 / gfx1250, wave32.
typedef __attribute__((ext_vector_type(2))) float v2f;
typedef __attribute__((ext_vector_type(8))) float v8f;

#define NPTS   4096
#define CHUNK  512
#define NCHUNK 8          // NPTS / CHUNK
#define ROWS   16         // rows (query points) per workgroup == one WMMA M tile
#define TPB    512        // 16 waves of 32

// Fused kernel: per workgroup of 16 waves:
//   phase 0: ||x_m||^2 into LDS
//   per 512-col chunk: WMMA 16x16x4_f32 Gram tiles -> pd chunk in LDS,
//                      then wave w slurps row w's chunk into registers
//   selection: 32 rounds of strictly-descending (val, -idx) argmax across the
//              register-resident row (128 vals/lane), wave32 shfl_xor reduce.
//   All lanes broadcast-load neighbor coords (uniform address); lane 0 writes.
__global__ __launch_bounds__(TPB)
void vn_knn_embed_kernel(const float* __restrict__ X, float* __restrict__ out)
{
    __shared__ float s_pd[ROWS * CHUNK];   // 32 KB
    __shared__ float s_xx[NPTS];           // 16 KB

    const int tid  = threadIdx.x;
    const int wave = tid >> 5;
    const int lane = tid & 31;
    const int b    = blockIdx.x >> 8;          // 256 row-blocks per batch
    const int rb   = blockIdx.x & 255;
    const int n0   = rb * ROWS;
    const float* __restrict__ xb = X + b * 3 * NPTS;   // [3][4096] row-major

    // ---- phase 0: squared norms ----
    for (int m = tid; m < NPTS; m += TPB) {
        float a0 = xb[m], a1 = xb[NPTS + m], a2 = xb[2 * NPTS + m];
        s_xx[m] = a0 * a0 + a1 * a1 + a2 * a2;
    }
    __syncthreads();

    // ---- A fragment (16 rows x K=4, K=3 zero-padded), shared by all tiles ----
    // 32-bit A 16x4 layout: lanes 0-15 hold K=0,1 ; lanes 16-31 hold K=2,3.
    const int hl     = lane >> 4;        // half-wave select (0: K=0,1  1: K=2,3)
    const int lcol   = lane & 15;
    const int kbase  = hl * 2 * NPTS;    // element offset of K=k0 plane
    // Zero the K=3 pad on the A side only (one-time cost). Because
    // D = sum_K A[M,K]*B[K,N], a zero A[M,3] makes B[3,N] don't-care, so the
    // per-tile B fragments below load unconditionally — no exec masking in
    // the hot loop.
    v2f afrag;
    afrag.x = xb[kbase + n0 + lcol];
    afrag.y = hl ? 0.0f : xb[NPTS + n0 + lcol];

    // per-lane row-||x||^2 terms, constant across all chunks/tiles
    float xxrow[8];
#pragma unroll
    for (int v = 0; v < 8; ++v)
        xxrow[v] = s_xx[n0 + v + 8 * hl];

    float vals[128];                     // full pd row, register resident

#pragma unroll
    for (int c = 0; c < NCHUNK; ++c) {
        const int cbase = c * CHUNK;
        // 32 column tiles per chunk, 2 per wave
#pragma unroll
        for (int t = 0; t < 2; ++t) {
            const int tile = wave * 2 + t;
            const int mt   = cbase + tile * 16;
            v2f bfrag;
            bfrag.x = xb[kbase + mt + lcol];
            // K=1 plane for lanes 0-15 (real), same plane lands in the K=3
            // slot for lanes 16-31 where it's multiplied by afrag.y == 0.
            bfrag.y = xb[NPTS + mt + lcol];
            v8f acc = {};
            acc = __builtin_amdgcn_wmma_f32_16x16x4_f32(
                false, afrag, false, bfrag, (short)0, acc, false, false);
            // D layout: lane 0-15 -> M=v, lane 16-31 -> M=v+8 ; N = lane&15
            const float xxc = s_xx[mt + lcol];
#pragma unroll
            for (int v = 0; v < 8; ++v) {
                const int row = v + 8 * hl;
                float pd = 2.0f * acc[v] - xxc - xxrow[v];
                s_pd[row * CHUNK + tile * 16 + lcol] = pd;
            }
        }
        __syncthreads();
        // wave w pulls row w's chunk: lane holds 16 contiguous columns
        const float4* rowp = (const float4*)&s_pd[wave * CHUNK + lane * 16];
#pragma unroll
        for (int q = 0; q < 4; ++q) {
            float4 f = rowp[q];
            vals[c * 16 + q * 4 + 0] = f.x;
            vals[c * 16 + q * 4 + 1] = f.y;
            vals[c * 16 + q * 4 + 2] = f.z;
            vals[c * 16 + q * 4 + 3] = f.w;
        }
        __syncthreads();
    }

    // ---- top-32 selection: strictly descending (val, -idx) argmax rounds ----
    float pv = __builtin_inff();
    int   pidx = -1;
    float sr0 = 0.f, sr1 = 0.f, sr2 = 0.f;   // real-coord running sum (k=32)
    float s80 = 0.f, s81 = 0.f, s82 = 0.f;   // snapshot at k=8
    float sc0 = 0.f, sc1 = 0.f, sc2 = 0.f;   // scrambled-view sum (k=16)

    for (int r = 0; r < 32; ++r) {
        float bv = -__builtin_inff();
        int   bidx = 0x7fffffff;
#pragma unroll
        for (int j = 0; j < 128; ++j) {
            float v  = vals[j];
            int  idx = (j >> 4) * CHUNK + lane * 16 + (j & 15);
            bool elig   = (v < pv) || (v == pv && idx > pidx);
            bool better = elig && ((v > bv) || (v == bv && idx < bidx));
            if (better) { bv = v; bidx = idx; }
        }
        // wave32 reduce, tie-break to lower index (matches jax top_k)
#pragma unroll
        for (int off = 16; off > 0; off >>= 1) {
            float ov = __shfl_xor(bv, off, 32);
            int   oi = __shfl_xor(bidx, off, 32);
            if (ov > bv || (ov == bv && oi < bidx)) { bv = ov; bidx = oi; }
        }
        pv = bv; pidx = bidx;
        // bidx is wave-uniform after the reduction: every lane issues the same
        // broadcast load (single request), accumulates redundantly, no exec
        // masking needed.
        sr0 += xb[bidx];
        sr1 += xb[NPTS + bidx];
        sr2 += xb[2 * NPTS + bidx];
        if (r < 16) {   // k=16 iteration gathers the reference's 3n+j view
            sc0 += xb[3 * bidx + 0];
            sc1 += xb[3 * bidx + 1];
            sc2 += xb[3 * bidx + 2];
        }
        if (r == 7) { s80 = sr0; s81 = sr1; s82 = sr2; }
    }

    // ---- write [B,4,3,N]: d=0 f32-mean, d=1 f16-mean(scrambled), d=2 f8-mean, d=3 self ----
    if (lane == 0) {
        const int n = n0 + wave;
        float* ob = out + b * (4 * 3 * NPTS);
        ob[(0 * 3 + 0) * NPTS + n] = sr0 * (1.0f / 32.0f);
        ob[(0 * 3 + 1) * NPTS + n] = sr1 * (1.0f / 32.0f);
        ob[(0 * 3 + 2) * NPTS + n] = sr2 * (1.0f / 32.0f);
        ob[(1 * 3 + 0) * NPTS + n] = sc0 * (1.0f / 16.0f);
        ob[(1 * 3 + 1) * NPTS + n] = sc1 * (1.0f / 16.0f);
        ob[(1 * 3 + 2) * NPTS + n] = sc2 * (1.0f / 16.0f);
        ob[(2 * 3 + 0) * NPTS + n] = s80 * (1.0f / 8.0f);
        ob[(2 * 3 + 1) * NPTS + n] = s81 * (1.0f / 8.0f);
        ob[(2 * 3 + 2) * NPTS + n] = s82 * (1.0f / 8.0f);
        ob[(3 * 3 + 0) * NPTS + n] = xb[n];
        ob[(3 * 3 + 1) * NPTS + n] = xb[NPTS + n];
        ob[(3 * 3 + 2) * NPTS + n] = xb[2 * NPTS + n];
    }
}

extern "C" void kernel_launch(void* const* d_in, const int* in_sizes, int n_in,
                              void* d_out, int out_size, void* d_ws, size_t ws_size,
                              hipStream_t stream)
{
    (void)in_sizes; (void)n_in; (void)d_ws; (void)ws_size; (void)out_size;
    const float* X = (const float*)d_in[0];   // [8,1,3,4096] fp32
    float* out = (float*)d_out;               // [8,4,3,4096,1] fp32
    const int B = 8;
    dim3 grid(B * (NPTS / ROWS));             // 2048 workgroups
    dim3 block(TPB);                          // 16 waves (wave32)
    vn_knn_embed_kernel<<<grid, block, 0, stream>>>(X, out);
}